// ShapeletsDistBlocks_7696581394834
// MI455X (gfx1250) — compile-verified
//
#include <hip/hip_runtime.h>

typedef __attribute__((ext_vector_type(2))) float v2f;
typedef __attribute__((ext_vector_type(8))) float v8f;

#define B_   32
#define C_   3
#define T_   2048
#define K_   128

__global__ void shapelet_init_out(float* out, int n) {
  int i = blockIdx.x * blockDim.x + threadIdx.x;
  if (i < n) out[i] = __int_as_float(0x7f800000);  // +inf
}

// One instantiation per shapelet length. grid = (wtiles, batches), block = 128 (4 waves).
// Each block: batch b, 16-window tile; 4 waves x 2 k-tiles cover 128 shapelets.
template <int S>
__global__ __launch_bounds__(128) void shapelet_min_kernel(
    const float* __restrict__ x,
    const float* __restrict__ sh,    // (C_, K_, S)
    float* __restrict__ out)         // pre-offset to this S's 128-column slab
{
  constexpr int W   = T_ - S + 1;    // valid window count
  constexpr int SEG = 15 + S;        // x elements needed per channel

  const int wtile = blockIdx.x;
  const int b     = blockIdx.y;
  const int w0    = wtile * 16;
  if (w0 >= W) return;               // block-uniform: EXEC stays all-ones

  __shared__ float xl[C_][SEG];      // x segment per channel (Hankel source)
  __shared__ float x2l[C_][16];      // window energies

  const int tid = threadIdx.x;
  for (int idx = tid; idx < C_ * SEG; idx += 128) {
    int c   = idx / SEG;
    int off = idx - c * SEG;
    int g   = w0 + off;
    if (g > T_ - 1) g = T_ - 1;      // clamp; invalid windows masked to +inf later
    xl[c][off] = x[((size_t)b * C_ + c) * T_ + g];
  }
  __syncthreads();
  if (tid < C_ * 16) {
    int c = tid >> 4, m = tid & 15;
    float s = 0.f;
    #pragma unroll
    for (int q = 0; q < S; ++q) { float v = xl[c][m + q]; s += v * v; }
    x2l[c][m] = s;
  }
  __syncthreads();

  const int wave = tid >> 5;
  const int lane = tid & 31;
  const int half = lane >> 4;        // lane half selects K-pair / M-half
  const int mn   = lane & 15;        // M for A-frag, N for B/C-frag
  const int koff = half * 2;         // K offset {0,1} vs {2,3}

  const float INF = __int_as_float(0x7f800000);

  #pragma unroll
  for (int it = 0; it < 2; ++it) {   // 4 waves x 2 tiles = 8 k-tiles = 128 shapelets
    const int kbase = (wave * 2 + it) * 16;
    v8f dsum = {};
    #pragma unroll
    for (int c = 0; c < C_; ++c) {
      const float* __restrict__ srow = sh + ((size_t)(c * K_ + kbase + mn)) * S;
      // ||shapelet||^2 for this lane's column
      float s2 = 0.f;
      #pragma unroll
      for (int q = 0; q < S; ++q) { float v = srow[q]; s2 += v * v; }

      v8f acc = {};
      #pragma unroll
      for (int kb = 0; kb < S; kb += 4) {
        v2f a, bb;
        // A (16x4 f32): lanes 0-15 hold K={kb,kb+1}, lanes 16-31 K={kb+2,kb+3}
        a.x = xl[c][mn + kb + koff];
        a.y = xl[c][mn + kb + koff + 1];
        // B (4x16 f32): row K per lane-half, N striped across lanes; 8B-aligned
        bb  = *(const v2f*)(srow + kb + koff);
        acc = __builtin_amdgcn_wmma_f32_16x16x4_f32(
            false, a, false, bb, (short)0, acc, false, false);
      }
      #pragma unroll
      for (int j = 0; j < 8; ++j) {
        int m = half * 8 + j;        // C/D layout: VGPR j holds M=j (+8 on upper half)
        float d2 = x2l[c][m] + s2 - 2.0f * acc[j];
        dsum[j] += sqrtf(fmaxf(d2, 1e-12f));
      }
    }
    // min over this lane's 8 windows, masking out-of-range ones
    float v = INF;
    #pragma unroll
    for (int j = 0; j < 8; ++j) {
      int m = half * 8 + j;
      v = fminf(v, (w0 + m < W) ? dsum[j] : INF);
    }
    v = fminf(v, __shfl_xor(v, 16, 32));   // fold the two M-halves
    if (half == 0) {
      // distances are >= 0, so IEEE float order == int order on the bit pattern
      int* o = (int*)(out + ((size_t)b * (3 * K_) + kbase + mn));
      atomicMin(o, __float_as_int(v));
    }
  }
}

extern "C" void kernel_launch(void* const* d_in, const int* in_sizes, int n_in,
                              void* d_out, int out_size, void* d_ws, size_t ws_size,
                              hipStream_t stream) {
  (void)in_sizes; (void)n_in; (void)d_ws; (void)ws_size;
  const float* x    = (const float*)d_in[0];
  const float* sh16 = (const float*)d_in[1];
  const float* sh32 = (const float*)d_in[2];
  const float* sh64 = (const float*)d_in[3];
  float* out = (float*)d_out;

  shapelet_init_out<<<(out_size + 255) / 256, 256, 0, stream>>>(out, out_size);

  dim3 block(128);
  // wtiles = ceil(W / 16) per S: W16=2033 -> 128, W32=2017 -> 127, W64=1985 -> 125
  shapelet_min_kernel<16><<<dim3(128, B_), block, 0, stream>>>(x, sh16, out + 0 * K_);
  shapelet_min_kernel<32><<<dim3(127, B_), block, 0, stream>>>(x, sh32, out + 1 * K_);
  shapelet_min_kernel<64><<<dim3(125, B_), block, 0, stream>>>(x, sh64, out + 2 * K_);
}